// G2R_83210696393549
// MI455X (gfx1250) — compile-verified
//
#include <hip/hip_runtime.h>

// ---------------- problem constants (match reference) ----------------
static constexpr int N_NODES = 50000;
static constexpr int N_EDGES = 1600000;
static constexpr int IN_DIM  = 64;
static constexpr int HID     = 128;
static constexpr int OUT_DIM = 64;
static constexpr int LPE     = 8;
static constexpr int NPERM   = 16;
static constexpr float LN_EPS = 1e-5f;

typedef __attribute__((ext_vector_type(2))) float v2f;
typedef __attribute__((ext_vector_type(8))) float v8f;

// ---------------- utility kernels ----------------
__global__ void zero_kernel(float* __restrict__ p, int n) {
    int tid = blockIdx.x * blockDim.x + threadIdx.x;
    if (tid < n) p[tid] = 0.0f;
}

// one thread per (edge, feature); d = 64 or 128, dshift = log2(d)
__global__ void scatter_add_kernel(const float* __restrict__ h,
                                   const int* __restrict__ src,
                                   const int* __restrict__ dst,
                                   float* __restrict__ agg,
                                   long long total, int dshift) {
    long long tid = (long long)blockIdx.x * blockDim.x + threadIdx.x;
    if (tid >= total) return;
    int e = (int)(tid >> dshift);
    int f = (int)(tid & ((1 << dshift) - 1));
    int s = src[e];
    int t = dst[e];
    float v = h[((long long)s << dshift) + f];
    atomicAdd(&agg[((long long)t << dshift) + f], v);
}

__device__ __forceinline__ void atomicMaxF(float* addr, float v) {
    // sign-aware integer trick: positives ordered as signed int, negatives as
    // decreasing unsigned -> max(float) == max(int) for v>=0, min(uint) for v<0
    if (v >= 0.0f) atomicMax((int*)addr, __float_as_int(v));
    else           atomicMin((unsigned int*)addr, __float_as_uint(v));
}

// one thread per (edge, perm-channel); NPERM = 16 fixed
__global__ void scatter_max_kernel(const float* __restrict__ c,
                                   const int* __restrict__ src,
                                   const int* __restrict__ dst,
                                   float* __restrict__ cnew,
                                   long long total) {
    long long tid = (long long)blockIdx.x * blockDim.x + threadIdx.x;
    if (tid >= total) return;
    int e = (int)(tid >> 4);
    int p = (int)(tid & 15);
    float v = c[(long long)src[e] * NPERM + p];
    atomicMaxF(&cnew[(long long)dst[e] * NPERM + p], v);
}

// c0[n,p] = perm[idx[n], p]
__global__ void perm_gather_kernel(const float* __restrict__ perm,
                                   const int* __restrict__ idx,
                                   float* __restrict__ c, int total) {
    int tid = blockIdx.x * blockDim.x + threadIdx.x;
    if (tid >= total) return;
    int n = tid >> 4, p = tid & 15;
    c[tid] = perm[(long long)idx[n] * NPERM + p];
}

// trans[n, p*LPE + l] = c[n,p]; optionally cnew = c (self-loop seed for max)
__global__ void pos_store_kernel(const float* __restrict__ c,
                                 float* __restrict__ trans,
                                 float* __restrict__ cnew,
                                 int total, int l) {
    int tid = blockIdx.x * blockDim.x + threadIdx.x;
    if (tid >= total) return;
    int n = tid >> 4, p = tid & 15;
    float v = c[tid];
    trans[(long long)n * (NPERM * LPE) + p * LPE + l] = v;
    if (cnew) cnew[tid] = v;
}

// ---------------- fused GEMM (+agg) + bias + LayerNorm + ReLU via f32 WMMA ----------------
// One wave computes a 16-row x DOUT tile: out = act(LN((in[+agg]) @ W + bias)).
// A-frag (16x4 f32): lanes 0-15 -> K={k0,k0+1}, lanes 16-31 -> K={k0+2,k0+3}, M = lane%16.
// B-frag (4x16 f32): same K split across lane halves, N = lane%16.
// C/D (16x16 f32): VGPR r -> row r (lanes 0-15) / row r+8 (lanes 16-31), col = lane%16.
template <int DIN, int DOUT, bool ADD_AGG, bool DO_LN>
__global__ __launch_bounds__(128) void gemm_ln_kernel(
    const float* __restrict__ in, const float* __restrict__ agg,
    const float* __restrict__ W, const float* __restrict__ bias,
    const float* __restrict__ gamma, const float* __restrict__ beta,
    float* __restrict__ out, int n_tiles) {
    constexpr int NT = DOUT / 16;
    int tile = blockIdx.x * (blockDim.x >> 5) + (threadIdx.x >> 5);
    if (tile >= n_tiles) return;        // wave-uniform: EXEC all-1s inside
    int lane = threadIdx.x & 31;
    int half = lane >> 4;
    int l15  = lane & 15;
    long long arow = (long long)(tile * 16 + l15) * DIN;

    v8f acc[NT] = {};
#pragma unroll
    for (int k0 = 0; k0 < DIN; k0 += 4) {
        int kk = k0 + half * 2;
        v2f a;
        if (ADD_AGG) {
            a.x = in[arow + kk]     + agg[arow + kk];
            a.y = in[arow + kk + 1] + agg[arow + kk + 1];
        } else {
            a.x = in[arow + kk];
            a.y = in[arow + kk + 1];
        }
#pragma unroll
        for (int t = 0; t < NT; t++) {
            v2f b;
            b.x = W[(long long)kk * DOUT + t * 16 + l15];
            b.y = W[(long long)(kk + 1) * DOUT + t * 16 + l15];
            acc[t] = __builtin_amdgcn_wmma_f32_16x16x4_f32(
                false, a, false, b, (short)0, acc[t], false, false);
        }
    }

    // bias (per-column)
#pragma unroll
    for (int t = 0; t < NT; t++) {
        float bc = bias[t * 16 + l15];
#pragma unroll
        for (int r = 0; r < 8; r++) acc[t][r] += bc;
    }

    if (DO_LN) {
        // LayerNorm across the full DOUT row, then ReLU.
        // Row r of this half lives across the 16 lanes of the half-wave:
        // width-16 xor shuffles reduce exactly that lane group.
#pragma unroll
        for (int r = 0; r < 8; r++) {
            float s = 0.0f, sq = 0.0f;
#pragma unroll
            for (int t = 0; t < NT; t++) { float v = acc[t][r]; s += v; sq += v * v; }
#pragma unroll
            for (int off = 8; off >= 1; off >>= 1) {
                s  += __shfl_xor(s,  off, 16);
                sq += __shfl_xor(sq, off, 16);
            }
            float mean = s * (1.0f / DOUT);
            float var  = sq * (1.0f / DOUT) - mean * mean;
            float rstd = rsqrtf(var + LN_EPS);
#pragma unroll
            for (int t = 0; t < NT; t++) {
                float g = gamma[t * 16 + l15];
                float b = beta[t * 16 + l15];
                float v = g * (acc[t][r] - mean) * rstd + b;
                acc[t][r] = v > 0.0f ? v : 0.0f;
            }
        }
    }

    // store D tiles
#pragma unroll
    for (int t = 0; t < NT; t++) {
#pragma unroll
        for (int r = 0; r < 8; r++) {
            int orow = tile * 16 + r + 8 * half;
            out[(long long)orow * DOUT + t * 16 + l15] = acc[t][r];
        }
    }
}

template <int DIN, int DOUT, bool ADD_AGG, bool DO_LN>
static void launch_gemm(const float* in, const float* agg, const float* W,
                        const float* bias, const float* gamma, const float* beta,
                        float* out, hipStream_t stream) {
    const int tiles = N_NODES / 16;              // 3125, exact
    dim3 grid((tiles + 3) / 4), block(128);      // 4 waves / block
    gemm_ln_kernel<DIN, DOUT, ADD_AGG, DO_LN>
        <<<grid, block, 0, stream>>>(in, agg, W, bias, gamma, beta, out, tiles);
}

// ---------------- driver ----------------
extern "C" void kernel_launch(void* const* d_in, const int* in_sizes, int n_in,
                              void* d_out, int out_size, void* d_ws, size_t ws_size,
                              hipStream_t stream) {
    (void)in_sizes; (void)n_in; (void)out_size; (void)ws_size;

    const float* x    = (const float*)d_in[0];
    const int*   ei   = (const int*)d_in[1];
    const int*   src  = ei;                 // edge_index[0]
    const int*   dst  = ei + N_EDGES;       // edge_index[1]
    const int*   idx  = (const int*)d_in[2];
    const float* gw0  = (const float*)d_in[3];
    const float* gb0  = (const float*)d_in[4];
    const float* gns0 = (const float*)d_in[5];
    const float* gnb0 = (const float*)d_in[6];
    const float* gw1  = (const float*)d_in[7];
    const float* gb1  = (const float*)d_in[8];
    const float* gns1 = (const float*)d_in[9];
    const float* gnb1 = (const float*)d_in[10];
    const float* gw2  = (const float*)d_in[11];
    const float* gb2  = (const float*)d_in[12];
    const float* gns2 = (const float*)d_in[13];
    const float* gnb2 = (const float*)d_in[14];
    const float* fc1w = (const float*)d_in[15];
    const float* fc1b = (const float*)d_in[16];
    const float* fcns = (const float*)d_in[17];
    const float* fcnb = (const float*)d_in[18];
    const float* fc2w = (const float*)d_in[19];
    const float* fc2b = (const float*)d_in[20];
    const float* posw1 = (const float*)d_in[21];
    const float* posb1 = (const float*)d_in[22];
    const float* posns = (const float*)d_in[23];
    const float* posnb = (const float*)d_in[24];
    const float* posw2 = (const float*)d_in[25];
    const float* posb2 = (const float*)d_in[26];
    const float* perm  = (const float*)d_in[27];

    float* regions = (float*)d_out;                     // [N, 64]
    float* pe      = regions + (long long)N_NODES * OUT_DIM;

    // workspace: buf0/buf1/buf2 = N*128 each, buf3 = N*256
    float* buf0 = (float*)d_ws;
    float* buf1 = buf0 + (long long)N_NODES * 128;
    float* buf2 = buf1 + (long long)N_NODES * 128;
    float* buf3 = buf2 + (long long)N_NODES * 128;

    const int THR = 256;
    auto blocks = [](long long n) { return (unsigned)((n + 255) / 256); };

    // ---------- GNN layer 0: agg over x (d=64), h1 = relu(LN((x+agg)@gw0+gb0)) ----------
    zero_kernel<<<blocks((long long)N_NODES * IN_DIM), THR, 0, stream>>>(buf0, N_NODES * IN_DIM);
    scatter_add_kernel<<<blocks((long long)N_EDGES * IN_DIM), THR, 0, stream>>>(
        x, src, dst, buf0, (long long)N_EDGES * IN_DIM, 6);
    launch_gemm<IN_DIM, HID, true, true>(x, buf0, gw0, gb0, gns0, gnb0, buf1, stream);

    // ---------- GNN layer 1 ----------
    zero_kernel<<<blocks((long long)N_NODES * HID), THR, 0, stream>>>(buf0, N_NODES * HID);
    scatter_add_kernel<<<blocks((long long)N_EDGES * HID), THR, 0, stream>>>(
        buf1, src, dst, buf0, (long long)N_EDGES * HID, 7);
    launch_gemm<HID, HID, true, true>(buf1, buf0, gw1, gb1, gns1, gnb1, buf2, stream);

    // ---------- GNN layer 2 ----------
    zero_kernel<<<blocks((long long)N_NODES * HID), THR, 0, stream>>>(buf0, N_NODES * HID);
    scatter_add_kernel<<<blocks((long long)N_EDGES * HID), THR, 0, stream>>>(
        buf2, src, dst, buf0, (long long)N_EDGES * HID, 7);
    launch_gemm<HID, HID, true, true>(buf2, buf0, gw2, gb2, gns2, gnb2, buf1, stream);

    // ---------- fc head: r = relu(LN(h@fc1w+fc1b)); regions = r@fc2w+fc2b ----------
    launch_gemm<HID, HID / 2, false, true>(buf1, nullptr, fc1w, fc1b, fcns, fcnb, buf2, stream);
    launch_gemm<HID / 2, OUT_DIM, false, false>(buf2, nullptr, fc2w, fc2b, nullptr, nullptr,
                                                regions, stream);

    // ---------- positional head: scatter-max propagation ----------
    // buf1 free after fc1 -> trans [N, NPERM*LPE]; buf0 -> c ping-pong
    float* trans = buf1;
    float* cA = buf0;
    float* cB = buf0 + (long long)N_NODES * NPERM;

    perm_gather_kernel<<<blocks((long long)N_NODES * NPERM), THR, 0, stream>>>(
        perm, idx, cA, N_NODES * NPERM);
    for (int l = 0; l < LPE - 1; ++l) {
        // record c at step l, seed cnew = c (self loops), then edge max
        pos_store_kernel<<<blocks((long long)N_NODES * NPERM), THR, 0, stream>>>(
            cA, trans, cB, N_NODES * NPERM, l);
        scatter_max_kernel<<<blocks((long long)N_EDGES * NPERM), THR, 0, stream>>>(
            cA, src, dst, cB, (long long)N_EDGES * NPERM);
        float* tmp = cA; cA = cB; cB = tmp;
    }
    pos_store_kernel<<<blocks((long long)N_NODES * NPERM), THR, 0, stream>>>(
        cA, trans, nullptr, N_NODES * NPERM, LPE - 1);

    // p = relu(LN(trans@posw1+posb1)); pe = p@posw2+posb2
    launch_gemm<NPERM * LPE, 2 * NPERM * LPE, false, true>(trans, nullptr, posw1, posb1,
                                                           posns, posnb, buf3, stream);
    launch_gemm<2 * NPERM * LPE, OUT_DIM, false, false>(buf3, nullptr, posw2, posb2,
                                                        nullptr, nullptr, pe, stream);
}